// TritonExpertDispatch_13572096655868
// MI455X (gfx1250) — compile-verified
//
#include <hip/hip_runtime.h>
#include <stdint.h>

// Problem constants (match the reference exactly)
#define N_TOK 16384
#define DIM   2048
#define NEXP  64
#define CAP   320          // int(16384/64 * 1.25)
#define TPB   256          // tokens per routing block
#define NBLK  (N_TOK / TPB) // 64

typedef float v4f __attribute__((ext_vector_type(4)));   // native vec4 for NT stores

// Workspace layout (in int32 units)
#define WS_HIST 0                       // [NBLK][NEXP] per-block expert counts
#define WS_BASE (WS_HIST + NBLK*NEXP)   // [NBLK][NEXP] exclusive prefix per expert
#define WS_CNT  (WS_BASE + NBLK*NEXP)   // [NEXP] total count per expert
#define WS_DEST (WS_CNT + NEXP)         // [N_TOK] dest slot (e*CAP+pos) or -1

// d_out layout (flat, in 4-byte units)
#define OFF_X 0LL                                   // dispatched_x: E*CAP*DIM f32
#define OFF_W (OFF_X + (long long)NEXP*CAP*DIM)     // combine_weights: E*CAP f32
#define OFF_I (OFF_W + (long long)NEXP*CAP)         // token_indices: E*CAP i32
#define OFF_D (OFF_I + (long long)NEXP*CAP)         // tokens_dropped: 1 i32

// ---------------------------------------------------------------------------
// Kernel 1: per-block expert histogram
// ---------------------------------------------------------------------------
__global__ __launch_bounds__(TPB) void k_hist(const int* __restrict__ ids,
                                              int* __restrict__ ws) {
    __shared__ int h[NEXP];
    const int t = threadIdx.x;
    if (t < NEXP) h[t] = 0;
    __syncthreads();
    const int e = ids[blockIdx.x * TPB + t];
    atomicAdd(&h[e], 1);
    __syncthreads();
    if (t < NEXP) ws[WS_HIST + blockIdx.x * NEXP + t] = h[t];
}

// ---------------------------------------------------------------------------
// Kernel 2: per-expert exclusive scan over blocks + totals + dropped count
// (single block of 64 threads; no atomics -> deterministic, no reset needed)
// ---------------------------------------------------------------------------
__global__ __launch_bounds__(NEXP) void k_scan(int* __restrict__ ws,
                                               int* __restrict__ dropped_out) {
    __shared__ int part[NEXP];
    const int e = threadIdx.x;   // 0..63
    int run = 0;
    for (int b = 0; b < NBLK; ++b) {
        ws[WS_BASE + b * NEXP + e] = run;
        run += ws[WS_HIST + b * NEXP + e];
    }
    ws[WS_CNT + e] = run;
    part[e] = (run > CAP) ? (run - CAP) : 0;
    __syncthreads();
    if (e == 0) {
        int s = 0;
        for (int i = 0; i < NEXP; ++i) s += part[i];
        *dropped_out = s;
    }
}

// ---------------------------------------------------------------------------
// Kernel 3: stable within-block rank -> dest slot; write scalar outputs
// ---------------------------------------------------------------------------
__global__ __launch_bounds__(TPB) void k_rank(const int* __restrict__ ids,
                                              const float* __restrict__ wts,
                                              int* __restrict__ ws,
                                              float* __restrict__ cw,
                                              int* __restrict__ ti) {
    __shared__ int sid[TPB];
    const int t = threadIdx.x;
    const int b = blockIdx.x;
    const int gt = b * TPB + t;
    const int e = ids[gt];
    sid[t] = e;
    __syncthreads();
    int r = 0;
    for (int j = 0; j < t; ++j) r += (sid[j] == e) ? 1 : 0;   // token-order stable
    const int pos = ws[WS_BASE + b * NEXP + e] + r;
    const int dst = (pos < CAP) ? (e * CAP + pos) : -1;
    ws[WS_DEST + gt] = dst;
    if (dst >= 0) {
        cw[dst] = wts[gt];
        ti[dst] = gt;
    }
}

// ---------------------------------------------------------------------------
// Kernel 4: zero only the unfilled tail slots (avoids double-writing 131 MB).
// Streaming zeros -> non-temporal b128 stores (footprint > 192 MB L2).
// ---------------------------------------------------------------------------
__global__ __launch_bounds__(TPB) void k_tail(const int* __restrict__ ws,
                                              float* __restrict__ dx,
                                              float* __restrict__ cw,
                                              int* __restrict__ ti) {
    const int slot = blockIdx.x;            // 0 .. E*CAP-1
    const int e = slot / CAP;
    const int s = slot % CAP;
    int c = ws[WS_CNT + e];
    if (c > CAP) c = CAP;
    if (s < c) return;                      // slot is filled by the scatter
    v4f* row = reinterpret_cast<v4f*>(dx + (size_t)slot * DIM);
    const int t = threadIdx.x;
    const v4f z = {0.f, 0.f, 0.f, 0.f};
    __builtin_nontemporal_store(z, &row[t]);        // 512 vec4 per row
    __builtin_nontemporal_store(z, &row[t + TPB]);  // 2 per thread
    if (t == 0) { cw[slot] = 0.f; ti[slot] = -1; }
}

// ---------------------------------------------------------------------------
// Kernel 5: row scatter of x via CDNA5 async global->LDS->global B128 path,
// with NT temporal hints (each byte touched exactly once; don't pollute L2).
// Each lane loads 16B into LDS and stores the same 16B out; per-wave
// s_wait_asynccnt is the only sync needed (no cross-lane LDS dependency).
// S_ENDPGM does an implicit wait-idle, so no final wait is required.
// ---------------------------------------------------------------------------
__global__ __launch_bounds__(TPB) void k_copy(const float* __restrict__ x,
                                              const int* __restrict__ ws,
                                              float* __restrict__ dx) {
    const int dst = ws[WS_DEST + blockIdx.x];
    if (dst < 0) return;                    // dropped token (uniform per block)
    __shared__ float lbuf[DIM];             // 8 KB staging
    const float* src = x  + (size_t)blockIdx.x * DIM;
    float*       out = dx + (size_t)dst      * DIM;
    // Low 32 bits of a flat LDS address are the LDS byte offset (ISA 10.2).
    const uint32_t lds0 = (uint32_t)(uintptr_t)&lbuf[0];
    const int t = threadIdx.x;

    // 512 float4 per row -> 2 per thread. Issue all async loads, wait, store.
    #pragma unroll
    for (int i = 0; i < 2; ++i) {
        const int q = i * TPB + t;                 // float4 index 0..511
        const uint32_t lo = lds0 + (uint32_t)q * 16u;
        const float* g = src + q * 4;
        asm volatile("global_load_async_to_lds_b128 %0, %1, off th:TH_LOAD_NT"
                     :: "v"(lo), "v"(g) : "memory");
    }
    asm volatile("s_wait_asynccnt 0x0" ::: "memory");
    #pragma unroll
    for (int i = 0; i < 2; ++i) {
        const int q = i * TPB + t;
        const uint32_t lo = lds0 + (uint32_t)q * 16u;
        float* g = out + q * 4;
        asm volatile("global_store_async_from_lds_b128 %0, %1, off th:TH_STORE_NT"
                     :: "v"(g), "v"(lo) : "memory");
    }
    // implicit S_WAIT_IDLE at S_ENDPGM covers outstanding async stores
}

// ---------------------------------------------------------------------------
extern "C" void kernel_launch(void* const* d_in, const int* in_sizes, int n_in,
                              void* d_out, int out_size, void* d_ws, size_t ws_size,
                              hipStream_t stream) {
    const float* x   = (const float*)d_in[0];
    const int*   ids = (const int*)d_in[1];
    const float* wts = (const float*)d_in[2];

    float* out_f = (float*)d_out;
    float* dx = out_f + OFF_X;
    float* cw = out_f + OFF_W;
    int*   ti = (int*)d_out + OFF_I;
    int*   dr = (int*)d_out + OFF_D;

    int* ws = (int*)d_ws;

    k_hist<<<NBLK, TPB, 0, stream>>>(ids, ws);
    k_scan<<<1, NEXP, 0, stream>>>(ws, dr);
    k_rank<<<NBLK, TPB, 0, stream>>>(ids, wts, ws, cw, ti);
    k_tail<<<NEXP * CAP, TPB, 0, stream>>>(ws, dx, cw, ti);
    k_copy<<<N_TOK, TPB, 0, stream>>>(x, ws, dx);
}